// NNUEEvaluator_66889820668652
// MI455X (gfx1250) — compile-verified
//
#include <hip/hip_runtime.h>

typedef __attribute__((ext_vector_type(2))) float v2f;
typedef __attribute__((ext_vector_type(8))) float v8f;

#define B_POS  4096
#define ACCW   512
#define A_FEAT 32
#define FEAT_W 1024   // 2*ACCW

// ---------------------------------------------------------------------------
// Kernel 1: feature transformer.
// One block per (position, perspective): gather 32 rows of 512 f32 from w_ft,
// sum, +bias, ReLU. 128 threads * float4 = 512 floats. 512 MB of gathers total
// against an 88 MB table -> L2-resident on MI455X (192 MB L2).
// ---------------------------------------------------------------------------
__global__ __launch_bounds__(128) void nnue_ft_kernel(
    const long long* __restrict__ white_idx,
    const long long* __restrict__ black_idx,
    const float* __restrict__ w_ft,
    const float* __restrict__ b_ft,
    float* __restrict__ features)
{
    const int task  = blockIdx.x;          // 0 .. 2*B_POS-1
    const int b     = task >> 1;
    const int persp = task & 1;
    const long long* __restrict__ idxp =
        (persp ? black_idx : white_idx) + (size_t)b * A_FEAT;
    const int t = threadIdx.x;             // float4 column 0..127
    const float4* __restrict__ wft4 = (const float4*)w_ft;

    float4 acc = {0.f, 0.f, 0.f, 0.f};
#pragma unroll 8
    for (int a = 0; a < A_FEAT; ++a) {
        const long long idx = idxp[a];     // uniform -> s_load
        float4 r = wft4[(size_t)idx * (ACCW / 4) + t];
        acc.x += r.x; acc.y += r.y; acc.z += r.z; acc.w += r.w;
    }
    const float4 bias = ((const float4*)b_ft)[t];
    float4 o;
    o.x = fmaxf(acc.x + bias.x, 0.f);
    o.y = fmaxf(acc.y + bias.y, 0.f);
    o.z = fmaxf(acc.z + bias.z, 0.f);
    o.w = fmaxf(acc.w + bias.w, 0.f);
    ((float4*)features)[(size_t)b * (FEAT_W / 4) + persp * (ACCW / 4) + t] = o;
}

// ---------------------------------------------------------------------------
// Kernel 2: dense head with V_WMMA_F32_16X16X4_F32.
// 256 threads = 8 waves; each wave owns a 16-row tile of the batch.
// Layer1: [16,1024]@[1024,32] -> two 16x16 WMMA accumulators, K stepped by 4.
// x1/x2 tiles staged in per-wave LDS slices; layer3 is a 32-wide dot product.
// A-matrix f32 16x4 layout: lane = 16*(K>=2)+M, vgpr = K%2.
// B-matrix f32 4x16 layout: lane = 16*(K>=2)+N, vgpr = K%2.
// C/D: vgpr r holds (M = r + 8*(lane>=16), N = lane%16).
// ---------------------------------------------------------------------------
__global__ __launch_bounds__(256) void nnue_head_kernel(
    const float* __restrict__ features,   // [B, 1024]
    const float* __restrict__ w1,         // [1024, 32]
    const float* __restrict__ b1,         // [32]
    const float* __restrict__ w2,         // [32, 32]
    const float* __restrict__ b2,         // [32]
    const float* __restrict__ w3,         // [32]
    const float* __restrict__ b3,         // [1]
    float* __restrict__ values)           // [B]
{
    __shared__ float xbuf[8 * 16 * 32];   // per-wave 16x32 activation tile
    const int lane = threadIdx.x & 31;
    const int wave = threadIdx.x >> 5;
    const int r0   = (blockIdx.x * 8 + wave) * 16;
    float* __restrict__ xw = &xbuf[wave * 16 * 32];

    const int M  = lane & 15;             // A-matrix row / B-matrix col
    const int kh = lane >> 4;             // K-half select

    // ---- layer 1 -----------------------------------------------------------
    v8f c0 = {0.f, 0.f, 0.f, 0.f, 0.f, 0.f, 0.f, 0.f};
    v8f c1 = {0.f, 0.f, 0.f, 0.f, 0.f, 0.f, 0.f, 0.f};
    const float* __restrict__ arow = features + (size_t)(r0 + M) * FEAT_W;
    for (int kb = 0; kb < FEAT_W; kb += 4) {
        const int kk = kb + 2 * kh;
        v2f a;   a.x  = arow[kk];               a.y  = arow[kk + 1];
        v2f bA;  bA.x = w1[kk * 32 + M];        bA.y = w1[(kk + 1) * 32 + M];
        v2f bB;  bB.x = w1[kk * 32 + 16 + M];   bB.y = w1[(kk + 1) * 32 + 16 + M];
        c0 = __builtin_amdgcn_wmma_f32_16x16x4_f32(false, a, false, bA,
                                                   (short)0, c0, false, false);
        c1 = __builtin_amdgcn_wmma_f32_16x16x4_f32(false, a, false, bB,
                                                   (short)0, c1, false, false);
    }
    {
        const float bb0 = b1[M], bb1 = b1[M + 16];
#pragma unroll
        for (int r = 0; r < 8; ++r) {
            const int row = r + 8 * kh;
            xw[row * 32 + M]      = fmaxf(c0[r] + bb0, 0.f);
            xw[row * 32 + M + 16] = fmaxf(c1[r] + bb1, 0.f);
        }
    }
    __syncthreads();

    // ---- layer 2: [16,32]@[32,32] -----------------------------------------
    v8f d0 = {0.f, 0.f, 0.f, 0.f, 0.f, 0.f, 0.f, 0.f};
    v8f d1 = {0.f, 0.f, 0.f, 0.f, 0.f, 0.f, 0.f, 0.f};
#pragma unroll
    for (int kb = 0; kb < 32; kb += 4) {
        const int kk = kb + 2 * kh;
        v2f a;   a.x  = xw[M * 32 + kk];        a.y  = xw[M * 32 + kk + 1];
        v2f bA;  bA.x = w2[kk * 32 + M];        bA.y = w2[(kk + 1) * 32 + M];
        v2f bB;  bB.x = w2[kk * 32 + 16 + M];   bB.y = w2[(kk + 1) * 32 + 16 + M];
        d0 = __builtin_amdgcn_wmma_f32_16x16x4_f32(false, a, false, bA,
                                                   (short)0, d0, false, false);
        d1 = __builtin_amdgcn_wmma_f32_16x16x4_f32(false, a, false, bB,
                                                   (short)0, d1, false, false);
    }
    __syncthreads();   // all reads of x1 done before overwriting xw
    {
        const float bb0 = b2[M], bb1 = b2[M + 16];
#pragma unroll
        for (int r = 0; r < 8; ++r) {
            const int row = r + 8 * kh;
            xw[row * 32 + M]      = fmaxf(d0[r] + bb0, 0.f);
            xw[row * 32 + M + 16] = fmaxf(d1[r] + bb1, 0.f);
        }
    }
    __syncthreads();

    // ---- layer 3: [16,32]@[32,1], *100 ------------------------------------
    if (lane < 16) {
        float v = b3[0];
#pragma unroll
        for (int k = 0; k < 32; ++k) v += xw[lane * 32 + k] * w3[k];
        values[r0 + lane] = v * 100.f;
    }
}

extern "C" void kernel_launch(void* const* d_in, const int* in_sizes, int n_in,
                              void* d_out, int out_size, void* d_ws, size_t ws_size,
                              hipStream_t stream) {
    const long long* white_idx = (const long long*)d_in[0];
    const long long* black_idx = (const long long*)d_in[1];
    const float* w_ft = (const float*)d_in[2];
    const float* b_ft = (const float*)d_in[3];
    const float* w1   = (const float*)d_in[4];
    const float* b1   = (const float*)d_in[5];
    const float* w2   = (const float*)d_in[6];
    const float* b2   = (const float*)d_in[7];
    const float* w3   = (const float*)d_in[8];
    const float* b3   = (const float*)d_in[9];

    float* features = (float*)d_out;                          // [4096*1024]
    float* values   = (float*)d_out + (size_t)B_POS * FEAT_W; // [4096]

    nnue_ft_kernel<<<B_POS * 2, 128, 0, stream>>>(
        white_idx, black_idx, w_ft, b_ft, features);

    nnue_head_kernel<<<B_POS / 128, 256, 0, stream>>>(
        features, w1, b1, w2, b2, w3, b3, values);
}